// GeneralContactLoss_31198642438394
// MI455X (gfx1250) — compile-verified
//
#include <hip/hip_runtime.h>
#include <math.h>

// ---------------------------------------------------------------------------
// GeneralContactLoss for MI455X (gfx1250, wave32).
//   B = 8, Q = 778, NF = 1538 (+14 constant FACES_NEW) = 1552 = 97*16 tris.
//
//   Winding numbers, reformulated for WMMA:
//     ct_i.ct_j = v_i.v_j - (v_i+v_j).p + |p|^2      (affine in p)
//     num       = d_t - p.n_t,  n_t = axb+bxc+cxa,  d_t = a.(bxc)
//   => per 16x16 (point x triangle) tile: 7x V_WMMA_F32_16X16X4_F32 with
//      A = [p0,p1,p2,1], then elementwise 3 sqrt + atan2.
//
//   Pairwise dist: P = |x|^2 + |y|^2 - 2 x.y^T, Gram via the same K=4 WMMA;
//   (sqrt(clip(P,0))*100)^2 == clip(P,0)*1e4 -> no sqrt needed.
// ---------------------------------------------------------------------------

#define BQ     8
#define QQ     778
#define NF_NEW 14
#define NF_TOT 1552            // 1538 + 14, == 97 * 16
#define NCT    (NF_TOT / 16)   // 97 triangle column tiles

typedef __attribute__((ext_vector_type(2))) float v2f;
typedef __attribute__((ext_vector_type(8))) float v8f;

__device__ __constant__ int c_faces_new[NF_NEW * 3] = {
    92, 38, 234, 234, 38, 239, 38, 122, 239, 239, 122, 279,
    122, 118, 279, 279, 118, 215, 118, 117, 215, 215, 117, 214,
    117, 119, 214, 214, 119, 121, 119, 120, 121, 121, 120, 78,
    120, 108, 78, 78, 108, 79};

// ------------------------------- init --------------------------------------
__global__ void gcl_init_kernel(float* buf, int n) {
    int i = blockIdx.x * blockDim.x + threadIdx.x;
    if (i < n) buf[i] = __uint_as_float(0x7f800000u);  // +INF
}

// --------------------- triangle coefficient precompute ---------------------
// For each (dir, batch, triangle): 7 quantities x 4 coeffs against [p,1]:
//   q0 num : [-n_t, d_t]
//   q1 d01 : [-(a+b), a.b]   (+|p|^2 elementwise later)
//   q2 d12 : [-(b+c), b.c]
//   q3 d02 : [-(a+c), a.c]
//   q4 d00 : [-2a, |a|^2]
//   q5 d11 : [-2b, |b|^2]
//   q6 d22 : [-2c, |c|^2]
// dir 0: triangles from v2, original order (r_faces)  -> interior_v1
// dir 1: triangles from v1, verts 1,2 swapped (l_faces) -> interior_v2
// Layout: coef[(dir*B+b)*7*NF_TOT*4 + q*NF_TOT*4 + n*4 + k]
__global__ void gcl_tricoef_kernel(const float* __restrict__ v1,
                                   const float* __restrict__ v2,
                                   const int* __restrict__ faces, int nf,
                                   float* __restrict__ coef) {
    const int idx = blockIdx.x * blockDim.x + threadIdx.x;
    if (idx >= 2 * BQ * NF_TOT) return;
    const int n   = idx % NF_TOT;
    const int b   = (idx / NF_TOT) % BQ;
    const int dir = idx / (NF_TOT * BQ);

    int i0, i1, i2;
    if (n < nf) {
        i0 = faces[3 * n]; i1 = faces[3 * n + 1]; i2 = faces[3 * n + 2];
    } else {
        const int t = n - nf;
        i0 = c_faces_new[3 * t]; i1 = c_faces_new[3 * t + 1]; i2 = c_faces_new[3 * t + 2];
    }
    if (dir == 1) { int t = i1; i1 = i2; i2 = t; }

    const float* V = (dir == 0 ? v2 : v1) + b * QQ * 3;
    const float ax = V[i0 * 3], ay = V[i0 * 3 + 1], az = V[i0 * 3 + 2];
    const float bx = V[i1 * 3], by = V[i1 * 3 + 1], bz = V[i1 * 3 + 2];
    const float cx = V[i2 * 3], cy = V[i2 * 3 + 1], cz = V[i2 * 3 + 2];

    // n_t = bxc + cxa + axb ; d_t = a.(bxc)
    const float bcx = by * cz - bz * cy, bcy = bz * cx - bx * cz, bcz = bx * cy - by * cx;
    const float cax = cy * az - cz * ay, cay = cz * ax - cx * az, caz = cx * ay - cy * ax;
    const float abx = ay * bz - az * by, aby = az * bx - ax * bz, abz = ax * by - ay * bx;
    const float ntx = bcx + cax + abx, nty = bcy + cay + aby, ntz = bcz + caz + abz;
    const float dt  = ax * bcx + ay * bcy + az * bcz;

    float* cb = coef + (size_t)(dir * BQ + b) * 7 * NF_TOT * 4;
    const int o = n * 4;
    const int PQ = NF_TOT * 4;
    // q0: num
    cb[0 * PQ + o] = -ntx; cb[0 * PQ + o + 1] = -nty; cb[0 * PQ + o + 2] = -ntz; cb[0 * PQ + o + 3] = dt;
    // q1: d01
    cb[1 * PQ + o] = -(ax + bx); cb[1 * PQ + o + 1] = -(ay + by); cb[1 * PQ + o + 2] = -(az + bz);
    cb[1 * PQ + o + 3] = ax * bx + ay * by + az * bz;
    // q2: d12
    cb[2 * PQ + o] = -(bx + cx); cb[2 * PQ + o + 1] = -(by + cy); cb[2 * PQ + o + 2] = -(bz + cz);
    cb[2 * PQ + o + 3] = bx * cx + by * cy + bz * cz;
    // q3: d02
    cb[3 * PQ + o] = -(ax + cx); cb[3 * PQ + o + 1] = -(ay + cy); cb[3 * PQ + o + 2] = -(az + cz);
    cb[3 * PQ + o + 3] = ax * cx + ay * cy + az * cz;
    // q4: d00
    cb[4 * PQ + o] = -2.f * ax; cb[4 * PQ + o + 1] = -2.f * ay; cb[4 * PQ + o + 2] = -2.f * az;
    cb[4 * PQ + o + 3] = ax * ax + ay * ay + az * az;
    // q5: d11
    cb[5 * PQ + o] = -2.f * bx; cb[5 * PQ + o + 1] = -2.f * by; cb[5 * PQ + o + 2] = -2.f * bz;
    cb[5 * PQ + o + 3] = bx * bx + by * by + bz * bz;
    // q6: d22
    cb[6 * PQ + o] = -2.f * cx; cb[6 * PQ + o + 1] = -2.f * cy; cb[6 * PQ + o + 2] = -2.f * cz;
    cb[6 * PQ + o + 3] = cx * cx + cy * cy + cz * cz;
}

// ----------------------- winding numbers via WMMA --------------------------
// grid = (49 row tiles, B, 2 dirs), block = 32 (one wave).
// A (16x4): lanes 0-15 K=0,1 ; lanes 16-31 K=2,3 ; M = lane%16.
// B (4x16): same K split, N = lane%16.  D element r: (M=r+8*half, N=lane%16).
__global__ void gcl_winding_wmma_kernel(const float* __restrict__ v1,
                                        const float* __restrict__ v2,
                                        const float* __restrict__ coef,
                                        int* __restrict__ interior) {
    const int rt = blockIdx.x, b = blockIdx.y, dir = blockIdx.z;
    const int lane = threadIdx.x;
    const int half = lane >> 4, mr = lane & 15;
    const int m = rt * 16 + mr;

    const float* P = (dir == 0 ? v1 : v2) + b * QQ * 3;  // query points
    float p0 = 0.f, p1 = 0.f, p2 = 0.f, ones = 0.f;
    if (m < QQ) { p0 = P[m * 3]; p1 = P[m * 3 + 1]; p2 = P[m * 3 + 2]; ones = 1.f; }
    v2f amat;
    amat.x = half ? p2 : p0;      // K = 2*half
    amat.y = half ? ones : p1;    // K = 2*half+1  (K=3 column is the constant 1)

    __shared__ float spp[16];     // |p|^2 per local row
    if (lane < 16) spp[mr] = p0 * p0 + p1 * p1 + p2 * p2;
    __syncthreads();

    float angsum[8];
#pragma unroll
    for (int r = 0; r < 8; ++r) angsum[r] = 0.f;

    const float2* cf = (const float2*)(coef + (size_t)(dir * BQ + b) * 7 * NF_TOT * 4);

    for (int ct = 0; ct < NCT; ++ct) {
        const int n = ct * 16 + mr;          // always < NF_TOT (exact tiling)
        v2f bq[7];
#pragma unroll
        for (int q = 0; q < 7; ++q) {
            const float2 t = cf[(size_t)q * NF_TOT * 2 + n * 2 + half];
            bq[q].x = t.x; bq[q].y = t.y;
        }
        v8f acc[7];
#pragma unroll
        for (int q = 0; q < 7; ++q) {
            v8f z = {};
            acc[q] = __builtin_amdgcn_wmma_f32_16x16x4_f32(
                false, amat, false, bq[q], (short)0, z, false, false);
        }
#pragma unroll
        for (int r = 0; r < 8; ++r) {
            const float pp  = spp[r + half * 8];
            const float num = acc[0][r];
            const float d01 = acc[1][r] + pp;
            const float d12 = acc[2][r] + pp;
            const float d02 = acc[3][r] + pp;
            const float la  = sqrtf(fmaxf(acc[4][r] + pp, 0.f));
            const float lb  = sqrtf(fmaxf(acc[5][r] + pp, 0.f));
            const float lc  = sqrtf(fmaxf(acc[6][r] + pp, 0.f));
            const float den = la * lb * lc + d01 * lc + d02 * lb + d12 * la;
            angsum[r] += atan2f(num, den);
        }
    }

    // sum over the 16 lanes of each half (all N) -> row totals
#pragma unroll
    for (int r = 0; r < 8; ++r) {
#pragma unroll
        for (int off = 8; off >= 1; off >>= 1)
            angsum[r] += __shfl_xor(angsum[r], off, 32);
    }
    if (mr == 0) {
        const float inv2pi = 1.0f / (2.0f * 3.14159265358979323846f);
#pragma unroll
        for (int r = 0; r < 8; ++r) {
            const int M = rt * 16 + r + half * 8;
            if (M < QQ)
                interior[(dir * BQ + b) * QQ + M] = (angsum[r] * inv2pi >= 0.99f) ? 1 : 0;
        }
    }
}

// -------------------- pairwise-dist tiles via WMMA -------------------------
__global__ void gcl_dist_wmma_kernel(const float* __restrict__ X,
                                     const float* __restrict__ Y,
                                     float* __restrict__ min12,
                                     float* __restrict__ min21) {
    const int b    = blockIdx.y;
    const int rt   = blockIdx.x;
    const int lane = threadIdx.x;
    const int half = lane >> 4, mr = lane & 15;
    const int m    = rt * 16 + mr;
    const float FINF = __uint_as_float(0x7f800000u);

    const float* Xb = X + b * QQ * 3;
    const float* Yb = Y + b * QQ * 3;

    float x0 = 0.f, x1 = 0.f, x2 = 0.f;
    if (m < QQ) { x0 = Xb[m * 3]; x1 = Xb[m * 3 + 1]; x2 = Xb[m * 3 + 2]; }
    v2f amat;
    amat.x = half ? x2 : x0;
    amat.y = half ? 0.f : x1;

    __shared__ float sxx[16];
    if (lane < 16) sxx[mr] = x0 * x0 + x1 * x1 + x2 * x2;
    __syncthreads();

    float rowmin[8];
#pragma unroll
    for (int r = 0; r < 8; ++r) rowmin[r] = FINF;

    const int ntiles = (QQ + 15) / 16;
    for (int ct = 0; ct < ntiles; ++ct) {
        const int n = ct * 16 + mr;
        float y0 = 0.f, y1 = 0.f, y2 = 0.f;
        if (n < QQ) { y0 = Yb[n * 3]; y1 = Yb[n * 3 + 1]; y2 = Yb[n * 3 + 2]; }
        v2f bmat;
        bmat.x = half ? y2 : y0;
        bmat.y = half ? 0.f : y1;
        const float yy = y0 * y0 + y1 * y1 + y2 * y2;

        v8f acc = {};
        acc = __builtin_amdgcn_wmma_f32_16x16x4_f32(
            false, amat, false, bmat, (short)0, acc, false, false);

        float colmin = FINF;
#pragma unroll
        for (int r = 0; r < 8; ++r) {
            const int M = rt * 16 + r + half * 8;
            float Pd = sxx[r + half * 8] + yy - 2.0f * acc[r];
            Pd = fmaxf(Pd, 0.0f);
            Pd = (M < QQ && n < QQ) ? Pd : FINF;
            rowmin[r] = fminf(rowmin[r], Pd);
            colmin    = fminf(colmin, Pd);
        }
        colmin = fminf(colmin, __shfl_xor(colmin, 16, 32));
        if (lane < 16 && n < QQ)
            atomicMin(reinterpret_cast<unsigned int*>(&min21[b * QQ + n]),
                      __float_as_uint(colmin));
    }

#pragma unroll
    for (int r = 0; r < 8; ++r) {
#pragma unroll
        for (int off = 8; off >= 1; off >>= 1)
            rowmin[r] = fminf(rowmin[r], __shfl_xor(rowmin[r], off, 32));
    }
    if (mr == 0) {
#pragma unroll
        for (int r = 0; r < 8; ++r) {
            const int M = rt * 16 + r + half * 8;
            if (M < QQ) min12[b * QQ + M] = rowmin[r];
        }
    }
}

// ------------------------------ finalize -----------------------------------
__global__ void gcl_finalize_kernel(const float* __restrict__ min12,
                                    const float* __restrict__ min21,
                                    const int* __restrict__ in1,
                                    const int* __restrict__ in2,
                                    float* __restrict__ out) {
    __shared__ float ssum[256];
    __shared__ int   sa1[256], sa2[256];
    const int tid = threadIdx.x;
    float total = 0.0f;

    for (int b = 0; b < BQ; ++b) {
        float part = 0.0f;
        int a1 = 0, a2 = 0;
        for (int q = tid; q < QQ; q += 256) {
            const int i1 = in1[b * QQ + q];
            const int i2 = in2[b * QQ + q];
            part += min12[b * QQ + q] * 10000.0f * (float)i1;
            part += min21[b * QQ + q] * 10000.0f * (float)i2;
            a1 |= i1;
            a2 |= i2;
        }
        ssum[tid] = part; sa1[tid] = a1; sa2[tid] = a2;
        __syncthreads();
        for (int s = 128; s > 0; s >>= 1) {
            if (tid < s) {
                ssum[tid] += ssum[tid + s];
                sa1[tid]  |= sa1[tid + s];
                sa2[tid]  |= sa2[tid + s];
            }
            __syncthreads();
        }
        if (tid == 0 && sa1[0] && sa2[0]) total += ssum[0];
        __syncthreads();
    }
    if (tid == 0) out[0] = total / (float)BQ;
}

// ------------------------------ launcher -----------------------------------
extern "C" void kernel_launch(void* const* d_in, const int* in_sizes, int n_in,
                              void* d_out, int out_size, void* d_ws, size_t ws_size,
                              hipStream_t stream) {
    (void)n_in; (void)out_size; (void)ws_size;
    const float* v1    = (const float*)d_in[0];
    const float* v2    = (const float*)d_in[1];
    const int*   faces = (const int*)d_in[2];
    const int    nf    = in_sizes[2] / 3;     // 1538
    float*       out   = (float*)d_out;

    // workspace: interior[2][B][Q] int, min12/min21 [B][Q] f32, coef planes
    int*   intr = (int*)d_ws;                       // 2*B*Q ints
    int*   in1  = intr;
    int*   in2  = intr + BQ * QQ;
    float* m12  = (float*)(intr + 2 * BQ * QQ);
    float* m21  = m12 + BQ * QQ;
    float* coef = m21 + BQ * QQ;                    // 2*B*7*NF_TOT*4 floats (~2.8 MB)

    const int nmin = 2 * BQ * QQ;
    gcl_init_kernel<<<(nmin + 255) / 256, 256, 0, stream>>>(m12, nmin);

    const int ntri = 2 * BQ * NF_TOT;
    gcl_tricoef_kernel<<<(ntri + 255) / 256, 256, 0, stream>>>(v1, v2, faces, nf, coef);

    dim3 wgrid((QQ + 15) / 16, BQ, 2);
    gcl_winding_wmma_kernel<<<wgrid, 32, 0, stream>>>(v1, v2, coef, intr);

    dim3 dgrid((QQ + 15) / 16, BQ);
    gcl_dist_wmma_kernel<<<dgrid, 32, 0, stream>>>(v1, v2, m12, m21);

    gcl_finalize_kernel<<<1, 256, 0, stream>>>(m12, m21, in1, in2, out);
}